// SPH_CAL_23313082483686
// MI455X (gfx1250) — compile-verified
//
#include <hip/hip_runtime.h>
#include <hip/hip_bf16.h>

// ---------------------------------------------------------------------------
// Compile-time constant tables (double precision, matching reference float64
// tables, cast to float exactly like jnp.asarray(_CA, float32)).
// ---------------------------------------------------------------------------
constexpr double csqrt(double x) {
  double g = x > 1.0 ? x : 1.0;
  for (int i = 0; i < 64; ++i) g = 0.5 * (g + x / g);
  return g;
}

struct SphConsts {
  double ca[28];
  double cb[28];
  double p1[7];
  double p2[7];
};

constexpr SphConsts make_consts() {
  SphConsts c{};
  for (int l = 1; l < 7; ++l) {
    for (int m = 0; m < l; ++m) {
      const int p = l * (l + 1) / 2 + m;
      c.ca[p] = csqrt((4.0 * l * l - 1.0) / (double)(l * l - m * m));
      const double num = (double)((l - 1) * (l - 1) - m * m);
      const double den = 4.0 * (l - 1) * (l - 1) - 1.0;
      c.cb[p] = (num == 0.0) ? 0.0 : -csqrt(num / den);
    }
  }
  c.p1[0] = -csqrt(1.5);
  for (int l = 1; l < 7; ++l) c.p1[l] = -csqrt(1.0 + 0.5 / (double)l);
  for (int l = 0; l < 7; ++l) c.p2[l] = csqrt(2.0 * l + 3.0);
  return c;
}

constexpr SphConsts SC = make_consts();

constexpr double PI_ = 3.14159265358979323846264338327950288;
constexpr double SQ2PIREV = csqrt(0.5 / PI_);   // sqrt(1/(2pi))
constexpr double SQ2REV   = csqrt(0.5);
constexpr float C00  = (float)(SQ2PIREV * SQ2REV);              // Y00
constexpr float C1XY = (float)(-csqrt(1.5) * SQ2PIREV);         // p1[1]*sqrt(1/2pi)
constexpr float C1Z  = (float)(csqrt(3.0) * SQ2REV * SQ2PIREV); // p2[0]*...
constexpr float HC1f = (float)csqrt(15.0 / (4.0 * PI_));
constexpr float HC2f = (float)csqrt(5.0 / (16.0 * PI_));
constexpr float HC3f = (float)csqrt(15.0 / (16.0 * PI_));

// ---------------------------------------------------------------------------
// Kernel config
// ---------------------------------------------------------------------------
constexpr int TPB = 256;       // 8 waves (wave32)
constexpr int VEC = 4;         // points per thread -> b128 stores
constexpr int PPB = TPB * VEC; // 1024 points per block

typedef float f32x4 __attribute__((ext_vector_type(4)));
typedef unsigned int u32;
typedef unsigned long long u64;
typedef u32 v4u __attribute__((ext_vector_type(4)));
typedef int  v4i __attribute__((ext_vector_type(4)));
typedef int  v8i __attribute__((ext_vector_type(8)));

#if defined(__has_builtin)
#if __has_builtin(__builtin_amdgcn_tensor_load_to_lds) && __has_builtin(__builtin_amdgcn_s_wait_tensorcnt)
#define SPH_HAVE_TDM 1
#endif
#endif

__global__ __launch_bounds__(TPB) void sph_cal_kernel(
    const float* __restrict__ cart, float* __restrict__ out, int n) {
  // LDS tile: [x 0..1023][y 0..1023][z 0..1023] = 12 KB
  __shared__ __align__(16) float tile[3 * PPB];

  const long long base = (long long)blockIdx.x * (long long)PPB;
  const int t = threadIdx.x;

  // Flat (generic) address of the LDS tile. On AMDGPU this is
  // {SRC_SHARED_BASE, lds_offset32}: the low 32 bits are exactly the
  // wave-relative LDS byte offset the TDM descriptor wants. Crucially this
  // ptrtoint *captures* tile[], so the compiler can no longer prove the
  // array is never written (the TDM writes are otherwise invisible to AA).
  const u64 tile_flat = (u64)(uintptr_t)(const void*)&tile[0];

#if defined(SPH_HAVE_TDM)
  // --- Tensor Data Mover: one wave DMAs the 1024x3 fp32 tile into LDS. ---
  // Tensor = 2D [dim0 = n (points), dim1 = 3 (x,y,z rows)], stride0 = n.
  // Tile   = [1024 x 3] starting at column `base`; OOB columns read as 0.
  if (t < 32) {
    const u64 ga = (u64)(const void*)cart + (u64)base * 4ull;
    v4u g0;
    g0[0] = 1u;                               // count=1, user descriptor
    g0[1] = (u32)tile_flat;                   // lds_addr = offset of tile[]
    g0[2] = (u32)(ga & 0xFFFFFFFFull);        // global_addr[31:0]
    g0[3] = (u32)((ga >> 32) & 0x01FFFFFFull) // global_addr[56:32]
            | (2u << 30);                     // type = 2 ("image")
    v8i g1;
    g1[0] = (int)(2u << 16);                              // data_size = 4B
    g1[1] = (int)(((u32)n & 0xFFFFu) << 16);              // tensor_dim0[15:0]
    g1[2] = (int)((((u32)n >> 16) & 0xFFFFu) | (3u << 16)); // dim0[31:16] | tensor_dim1=3
    g1[3] = (int)((u32)PPB << 16);                        // tile_dim0 = 1024
    g1[4] = 3;                                            // tile_dim1 = 3
    g1[5] = n;                                            // tensor_dim0_stride[31:0]
    g1[6] = 0;
    g1[7] = 0;
    v4i g2 = {0, 0, 0, 0};
    v4i g3 = {0, 0, 0, 0};
#if defined(__clang_major__) && (__clang_major__ >= 23)
    v8i g4 = {0, 0, 0, 0, 0, 0, 0, 0};
    __builtin_amdgcn_tensor_load_to_lds(g0, g1, g2, g3, g4, 0);
#else
    __builtin_amdgcn_tensor_load_to_lds(g0, g1, g2, g3, 0);
#endif
    __builtin_amdgcn_s_wait_tensorcnt(0);
  }
  __syncthreads();
#else
  // Fallback: cooperative coalesced load of the tile.
  for (int i = t; i < 3 * PPB; i += TPB) {
    const int row = i / PPB;
    const int col = i - row * PPB;
    const long long g = base + col;
    tile[i] = (g < (long long)n) ? cart[(size_t)row * (size_t)n + (size_t)g] : 0.0f;
  }
  __syncthreads();
#endif

  // Opaque may-writer of tile[] (whose address is captured above), ordered
  // between the DMA/barrier and the reads: forces the tile loads to
  // materialize as real ds_load instructions instead of folding to undef.
  {
    u64 tf = tile_flat;
    asm volatile("" : "+v"(tf) : : "memory");
  }

  const long long i0 = base + (long long)t * VEC;

  float X[VEC], Y[VEC], Z[VEC], D2[VEC];
  float s[VEC][49];

  // Direct scalar indexing of the __shared__ array keeps the accesses in
  // address space 3 (-> ds_load_b128 after vectorization), instead of the
  // flat-load path a generic-pointer cast would produce.
#pragma unroll
  for (int v = 0; v < VEC; ++v) X[v] = tile[t * VEC + v];
#pragma unroll
  for (int v = 0; v < VEC; ++v) Y[v] = tile[PPB + t * VEC + v];
#pragma unroll
  for (int v = 0; v < VEC; ++v) Z[v] = tile[2 * PPB + t * VEC + v];

  const bool fullvec = ((i0 + VEC) <= (long long)n) && ((n & (VEC - 1)) == 0);

  // Store one output row (index k) for this thread's 4 points, non-temporal.
  auto stb = [&](int k) {
    if (fullvec) {
      f32x4 o = {s[0][k], s[1][k], s[2][k], s[3][k]};
      __builtin_nontemporal_store(o, (f32x4*)(out + (size_t)k * (size_t)n + (size_t)i0));
    } else {
#pragma unroll
      for (int v = 0; v < VEC; ++v) {
        if (i0 + v < (long long)n) {
          __builtin_nontemporal_store(s[v][k], out + (size_t)k * (size_t)n + (size_t)(i0 + v));
        }
      }
    }
  };

  // Bands l = 0, 1, 2 (hard-coded, matching reference).
#pragma unroll
  for (int v = 0; v < VEC; ++v) {
    const float x = X[v], y = Y[v], z = Z[v];
    const float q = x * x + y * y + z * z;
    D2[v] = q;
    s[v][0] = C00;
    s[v][1] = C1XY * y;
    s[v][2] = C1Z * z;
    s[v][3] = C1XY * x;
    s[v][4] = HC1f * x * y;
    s[v][5] = -HC1f * y * z;
    s[v][6] = HC2f * (3.0f * z * z - q);
    s[v][7] = -HC1f * x * z;
    s[v][8] = HC3f * (x * x - y * y);
  }
#pragma unroll
  for (int k = 0; k < 9; ++k) stb(k);

  // Bands l = 3..6 via the two-term recurrence; store each band as computed
  // so only bands (l-1, l-2) stay live in VGPRs.
#pragma unroll
  for (int l = 3; l < 7; ++l) {
    const int c = l * l + l;             // center of band l
    const int cp = (l - 1) * l;          // center of band l-1
    const int cpp = (l - 2) * (l - 1);   // center of band l-2
#pragma unroll
    for (int v = 0; v < VEC; ++v) {
      const float x = X[v], y = Y[v], z = Z[v], q = D2[v];
#pragma unroll
      for (int m = 0; m < l - 1; ++m) {
        const float a = (float)SC.ca[l * (l + 1) / 2 + m];
        const float b = (float)SC.cb[l * (l + 1) / 2 + m];
        s[v][c + m] = a * (z * s[v][cp + m] + b * q * s[v][cpp + m]);
        if (m > 0) {
          s[v][c - m] = a * (z * s[v][cp - m] + b * q * s[v][cpp - m]);
        }
      }
      const float tp = s[v][cp + (l - 1)];
      const float tn = s[v][cp - (l - 1)];
      const float P2v = (float)SC.p2[l - 1];
      const float P1v = (float)SC.p1[l];
      s[v][c + (l - 1)] = P2v * z * tp;
      s[v][c - (l - 1)] = P2v * z * tn;
      s[v][c + l] = P1v * (x * tp - y * tn);
      s[v][c - l] = P1v * (x * tn + y * tp);
    }
#pragma unroll
    for (int k = l * l; k < (l + 1) * (l + 1); ++k) stb(k);
  }
}

// ---------------------------------------------------------------------------
// Host launch
// ---------------------------------------------------------------------------
extern "C" void kernel_launch(void* const* d_in, const int* in_sizes, int n_in,
                              void* d_out, int out_size, void* d_ws, size_t ws_size,
                              hipStream_t stream) {
  (void)n_in; (void)out_size; (void)d_ws; (void)ws_size;
  const float* cart = (const float*)d_in[0];
  float* out = (float*)d_out;
  const int n = in_sizes[0] / 3;  // cart is (3, N) flat
  const int blocks = (n + PPB - 1) / PPB;
  sph_cal_kernel<<<blocks, TPB, 0, stream>>>(cart, out, n);
}